// LineGCN2_69217692942493
// MI455X (gfx1250) — compile-verified
//
#include <hip/hip_runtime.h>
#include <hip/hip_bf16.h>
#include <math.h>

typedef __attribute__((ext_vector_type(2))) float v2f;
typedef __attribute__((ext_vector_type(8))) float v8f;

#define HID 256
#define ADJ_STRIDE 32

// ---------------------------------------------------------------------------
// GEMM  C[M,Nn] = A[M,K] @ B[K,Nn]  via V_WMMA_F32_16X16X4_F32 (fp32 exact).
// One wave per 16x16 output tile. M % 16 == 0, Nn % 16 == 0, K % 4 == 0.
// A 16x4 layout : lanes 0-15 -> M=lane, K = k0+{0,1}; lanes 16-31 -> K = k0+{2,3}
// B 4x16 layout : lanes 0-15 -> N=lane, K = k0+{0,1}; lanes 16-31 -> K = k0+{2,3}
// C 16x16      : vgpr r -> M = r + (lane>=16 ? 8 : 0), N = lane&15
// ---------------------------------------------------------------------------
__global__ __launch_bounds__(256)
void gemm_wmma_f32(const float* __restrict__ A, const float* __restrict__ B,
                   float* __restrict__ C, int M, int K, int Nn) {
    const int wave = blockIdx.x * (blockDim.x >> 5) + (threadIdx.x >> 5);
    const int lane = threadIdx.x & 31;
    const int ntiles = Nn >> 4;
    const int totTiles = (M >> 4) * ntiles;
    if (wave >= totTiles) return;                 // wave-uniform exit
    const int mt = wave / ntiles;
    const int nt = wave - mt * ntiles;
    const int row   = mt * 16 + (lane & 15);      // A row for this lane
    const int col   = nt * 16 + (lane & 15);      // B/C column for this lane
    const int khalf = (lane >> 4) << 1;           // 0 for lanes 0-15, 2 for 16-31

    v8f acc = {};
    const float* Arow = A + (size_t)row * K + khalf;
    #pragma unroll 4
    for (int k0 = 0; k0 < K; k0 += 4) {
        v2f a = *(const v2f*)(Arow + k0);         // A[row][k0+khalf .. +1]
        v2f b;
        b.x = B[(size_t)(k0 + khalf)     * Nn + col];
        b.y = B[(size_t)(k0 + khalf + 1) * Nn + col];
        acc = __builtin_amdgcn_wmma_f32_16x16x4_f32(
            /*neg_a=*/false, a, /*neg_b=*/false, b,
            /*c_mod=*/(short)0, acc, /*reuse_a=*/false, /*reuse_b=*/false);
    }
    const int rbase = mt * 16 + ((lane >> 4) << 3);
    #pragma unroll
    for (int r = 0; r < 8; ++r)
        C[(size_t)(rbase + r) * Nn + col] = acc[r];
}

// ------------------------- graph prep --------------------------------------
__global__ void build_adj_k(const int* __restrict__ src, const int* __restrict__ dst,
                            int E, int* __restrict__ cnt, int* __restrict__ adj) {
    int e = blockIdx.x * blockDim.x + threadIdx.x;
    if (e >= E) return;
    int v = dst[e];
    int slot = atomicAdd(&cnt[v], 1);
    if (slot < ADJ_STRIDE) adj[v * ADJ_STRIDE + slot] = src[e];
}

__global__ void make_dinv_k(const int* __restrict__ cnt, float* __restrict__ dinv, int n) {
    int i = blockIdx.x * blockDim.x + threadIdx.x;
    if (i < n) dinv[i] = rsqrtf((float)(cnt[i] + 1));   // +1 self loop
}

// GCN aggregation: O[v] = dinv[v]*( sum_u dinv[u]*T[u] + dinv[v]*T[v] ) + bias
__global__ __launch_bounds__(HID)
void gcn_aggregate_k(const float* __restrict__ T, const int* __restrict__ cnt,
                     const int* __restrict__ adj, const float* __restrict__ dinv,
                     const float* __restrict__ bias, float* __restrict__ O,
                     int doRelu) {
    const int v = blockIdx.x;
    const int f = threadIdx.x;
    const float dv = dinv[v];
    const int deg = min(cnt[v], ADJ_STRIDE);
    float sum = dv * T[(size_t)v * HID + f];     // self-loop term (dinv[v]*x_v)
    for (int k = 0; k < deg; ++k) {
        int u = adj[v * ADJ_STRIDE + k];
        sum += dinv[u] * T[(size_t)u * HID + f];
    }
    float r = dv * sum + bias[f];
    if (doRelu) r = fmaxf(r, 0.0f);
    O[(size_t)v * HID + f] = r;
}

// ------------------------- batch norm --------------------------------------
__global__ __launch_bounds__(HID)
void bn_partial_k(const float* __restrict__ X, int N,
                  float* __restrict__ sum, float* __restrict__ sumsq) {
    const int f = threadIdx.x;
    float s = 0.f, ss = 0.f;
    for (int r = blockIdx.x; r < N; r += gridDim.x) {
        float x = X[(size_t)r * HID + f];
        s += x; ss += x * x;
    }
    atomicAdd(&sum[f], s);
    atomicAdd(&sumsq[f], ss);
}

__global__ void bn_final_k(const float* __restrict__ sum, const float* __restrict__ sumsq,
                           const float* __restrict__ gamma, const float* __restrict__ beta,
                           float* __restrict__ scale, float* __restrict__ shift, int N) {
    int f = threadIdx.x;
    float mu  = sum[f] / (float)N;
    float var = sumsq[f] / (float)N - mu * mu;
    float rs  = rsqrtf(var + 1e-5f);
    float sc  = gamma[f] * rs;
    scale[f] = sc;
    shift[f] = beta[f] - mu * sc;
}

__global__ void bn_apply_relu_k(float* __restrict__ X, const float* __restrict__ scale,
                                const float* __restrict__ shift, int total) {
    int i = blockIdx.x * blockDim.x + threadIdx.x;
    if (i >= total) return;
    int f = i & (HID - 1);
    X[i] = fmaxf(X[i] * scale[f] + shift[f], 0.0f);
}

// ---------------- line-graph head: s[e] = h[src]·Wl[:256] + h[dst]·Wl[256:] -
__global__ __launch_bounds__(256)
void node_dots_k(const float* __restrict__ H, const float* __restrict__ Wl,
                 float* __restrict__ av, float* __restrict__ bv, int N) {
    int node = blockIdx.x * 8 + (threadIdx.x >> 5);
    int lane = threadIdx.x & 31;
    if (node >= N) return;
    float a = 0.f, b = 0.f;
    for (int f = lane; f < HID; f += 32) {
        float h = H[(size_t)node * HID + f];
        a += h * Wl[f];
        b += h * Wl[HID + f];
    }
    #pragma unroll
    for (int off = 16; off; off >>= 1) {
        a += __shfl_down(a, off, 32);
        b += __shfl_down(b, off, 32);
    }
    if (lane == 0) { av[node] = a; bv[node] = b; }
}

__global__ void edge_score_k(const int* __restrict__ src, const int* __restrict__ dst,
                             const float* __restrict__ av, const float* __restrict__ bv,
                             float* __restrict__ s, int E) {
    int e = blockIdx.x * blockDim.x + threadIdx.x;
    if (e < E) s[e] = av[src[e]] + bv[dst[e]];
}

__global__ void count_dst_k(const int* __restrict__ ldst, int LE, int* __restrict__ cnt) {
    int j = blockIdx.x * blockDim.x + threadIdx.x;
    if (j < LE) atomicAdd(&cnt[ldst[j]], 1);
}

__global__ void line_scatter_k(const int* __restrict__ lsrc, const int* __restrict__ ldst,
                               const float* __restrict__ s, const float* __restrict__ dl,
                               float* __restrict__ acc, int LE) {
    int j = blockIdx.x * blockDim.x + threadIdx.x;
    if (j >= LE) return;
    int u = lsrc[j];
    atomicAdd(&acc[ldst[j]], dl[u] * s[u]);
}

__global__ void finalize_k(const float* __restrict__ acc, const float* __restrict__ s,
                           const float* __restrict__ dl, const float* __restrict__ bl,
                           float* __restrict__ out, int E) {
    int e = blockIdx.x * blockDim.x + threadIdx.x;
    if (e >= E) return;
    float d = dl[e];
    float t = d * acc[e] + d * d * s[e] + bl[0];
    out[e] = 1.0f / (1.0f + expf(-t));
}

// ---------------------------------------------------------------------------
extern "C" void kernel_launch(void* const* d_in, const int* in_sizes, int n_in,
                              void* d_out, int out_size, void* d_ws, size_t ws_size,
                              hipStream_t stream) {
    const float* x      = (const float*)d_in[0];
    const float* W1     = (const float*)d_in[1];
    const float* b1     = (const float*)d_in[2];
    const float* gamma1 = (const float*)d_in[3];
    const float* beta1  = (const float*)d_in[4];
    const float* W2     = (const float*)d_in[5];
    const float* b2     = (const float*)d_in[6];
    const float* Wl     = (const float*)d_in[7];
    const float* bl     = (const float*)d_in[8];
    const int*   ei     = (const int*)d_in[9];
    const int*   lei    = (const int*)d_in[10];

    const int N  = in_sizes[0] / 128;     // 20000
    const int K1 = in_sizes[1] / HID;     // 128
    const int E  = in_sizes[9] / 2;       // 320000
    const int LE = in_sizes[10] / 2;      // 5120000
    const int* src  = ei;
    const int* dst  = ei + E;
    const int* lsrc = lei;
    const int* ldst = lei + LE;
    float* out = (float*)d_out;

    // workspace carve-up (256B aligned)
    char* w = (char*)d_ws;
    auto carve = [&](size_t bytes) -> void* {
        void* p = (void*)w;
        w += (bytes + 255) & ~(size_t)255;
        return p;
    };
    float* bufA  = (float*)carve((size_t)N * HID * 4);
    float* bufB  = (float*)carve((size_t)N * HID * 4);
    float* dinvN = (float*)carve((size_t)N * 4);
    float* dinvL = (float*)carve((size_t)E * 4);
    float* svec  = (float*)carve((size_t)E * 4);
    float* accv  = (float*)carve((size_t)E * 4);
    float* avec  = (float*)carve((size_t)N * 4);
    float* bvec  = (float*)carve((size_t)N * 4);
    float* colsum= (float*)carve(HID * 4);
    float* colss = (float*)carve(HID * 4);
    float* bnsc  = (float*)carve(HID * 4);
    float* bnsh  = (float*)carve(HID * 4);
    int*   cnt   = (int*)carve((size_t)(E > N ? E : N) * 4);
    int*   adj   = (int*)carve((size_t)N * ADJ_STRIDE * 4);

    const int T256 = 256;
    // 1) node-graph adjacency + dinv
    hipMemsetAsync(cnt, 0, (size_t)N * 4, stream);
    build_adj_k<<<(E + T256 - 1) / T256, T256, 0, stream>>>(src, dst, E, cnt, adj);
    make_dinv_k<<<(N + T256 - 1) / T256, T256, 0, stream>>>(cnt, dinvN, N);

    // 2) GEMM1: bufA = x @ W1   (M=N, K=K1, Nn=HID)
    {
        int tiles = (N >> 4) * (HID >> 4);
        gemm_wmma_f32<<<(tiles + 7) / 8, 256, 0, stream>>>(x, W1, bufA, N, K1, HID);
    }
    // 3) aggregate + b1 (no relu yet)
    gcn_aggregate_k<<<N, HID, 0, stream>>>(bufA, cnt, adj, dinvN, b1, bufB, 0);

    // 4) batch norm + relu (in place on bufB)
    hipMemsetAsync(colsum, 0, HID * 4, stream);
    hipMemsetAsync(colss,  0, HID * 4, stream);
    bn_partial_k<<<256, HID, 0, stream>>>(bufB, N, colsum, colss);
    bn_final_k<<<1, HID, 0, stream>>>(colsum, colss, gamma1, beta1, bnsc, bnsh, N);
    bn_apply_relu_k<<<((size_t)N * HID + T256 - 1) / T256, T256, 0, stream>>>(bufB, bnsc, bnsh, N * HID);

    // 5) GEMM2: bufA = bufB @ W2   (K=HID)
    {
        int tiles = (N >> 4) * (HID >> 4);
        gemm_wmma_f32<<<(tiles + 7) / 8, 256, 0, stream>>>(bufB, W2, bufA, N, HID, HID);
    }
    // 6) aggregate + b2 + relu -> bufB = h2
    gcn_aggregate_k<<<N, HID, 0, stream>>>(bufA, cnt, adj, dinvN, b2, bufB, 1);

    // 7) per-node dots with Wl halves
    node_dots_k<<<(N + 7) / 8, 256, 0, stream>>>(bufB, Wl, avec, bvec, N);
    // 8) per-edge (line-node) scalar feature
    edge_score_k<<<(E + T256 - 1) / T256, T256, 0, stream>>>(src, dst, avec, bvec, svec, E);

    // 9) line-graph degrees -> dinvL
    hipMemsetAsync(cnt, 0, (size_t)E * 4, stream);
    count_dst_k<<<(LE + T256 - 1) / T256, T256, 0, stream>>>(ldst, LE, cnt);
    make_dinv_k<<<(E + T256 - 1) / T256, T256, 0, stream>>>(cnt, dinvL, E);

    // 10) scatter-aggregate over line edges
    hipMemsetAsync(accv, 0, (size_t)E * 4, stream);
    line_scatter_k<<<(LE + T256 - 1) / T256, T256, 0, stream>>>(lsrc, ldst, svec, dinvL, accv, LE);

    // 11) finalize: norm + self loop + bias + sigmoid
    finalize_k<<<(E + T256 - 1) / T256, T256, 0, stream>>>(accv, svec, dinvL, bl, out, E);
}